// RecurrentGCN_20409684591099
// MI455X (gfx1250) — compile-verified
//
#include <hip/hip_runtime.h>
#include <math.h>

#define PP  12
#define HD  100

typedef __attribute__((ext_vector_type(2))) float v2f;
typedef __attribute__((ext_vector_type(8))) float v8f;

// ---------------------------------------------------------------------------
// K1: zero agg, set deg = 1.0 (self-loop weight folded into degree init)
// ---------------------------------------------------------------------------
__global__ void k_init(float* __restrict__ agg, float* __restrict__ deg, int N) {
    int i = blockIdx.x * blockDim.x + threadIdx.x;
    if (i < N * PP) agg[i] = 0.0f;
    if (i < N)      deg[i] = 1.0f;
}

// ---------------------------------------------------------------------------
// K2: single-wave precompute.
//   For g in {z,h}: fold weights with V_WMMA_F32_16X16X4_F32:
//     A(16x4): row0 = W_g chunk, row1 = b_g chunk, rows 2..15 = 0
//     B(4x16): Wl_g[k, j0+n]  (top HID rows only; bottom half multiplies H0=0)
//     D row0 -> u_g tile, D row1 (+ bl_g) -> c_g tile
//   Also: probs = softmax(att) via shuffle reductions.
// 32 threads; EXEC is all-1s at every WMMA (selects, no divergence).
// ---------------------------------------------------------------------------
__global__ void k_precompute(const float* __restrict__ Wz, const float* __restrict__ bz,
                             const float* __restrict__ Wh, const float* __restrict__ bh,
                             const float* __restrict__ Wlz, const float* __restrict__ blz,
                             const float* __restrict__ Wlh, const float* __restrict__ blh,
                             const float* __restrict__ att,
                             float* __restrict__ uz, float* __restrict__ cz,
                             float* __restrict__ uh, float* __restrict__ ch,
                             float* __restrict__ probs) {
    const int lane = threadIdx.x;        // 0..31
    const int m    = lane & 15;          // A-matrix row / D column index
    const int khi  = (lane < 16) ? 0 : 2;

    for (int g = 0; g < 2; ++g) {
        const float* Wg = g ? Wh  : Wz;
        const float* bg = g ? bh  : bz;
        const float* Wl = g ? Wlh : Wlz;
        const float* bl = g ? blh : blz;
        float*       uo = g ? uh  : uz;
        float*       co = g ? ch  : cz;

        for (int j0 = 0; j0 < HD; j0 += 16) {   // 7 N-tiles (covers j=0..111)
            const int n = j0 + m;               // B/D column this lane owns
            v8f acc = {};
            for (int k0 = 0; k0 < HD; k0 += 4) {  // 25 K-chunks, HD%4==0
                const int ka = k0 + khi;          // always < HD
                const int kb = ka + 1;            // always < HD
                // A layout: lane L -> A[L&15, k0 + (L<16 ? {0,1} : {2,3})]
                float w0 = Wg[ka], w1 = Wg[kb];
                float c0 = bg[ka], c1 = bg[kb];
                v2f a;
                a.x = (m == 0) ? w0 : ((m == 1) ? c0 : 0.0f);
                a.y = (m == 0) ? w1 : ((m == 1) ? c1 : 0.0f);
                // B layout: lane L -> B[k0 + (L<16 ? {0,1} : {2,3}), L&15]
                // indices stay inside Wl (max 99*100+111 < 2*HD*HD)
                v2f b;
                b.x = Wl[ka * HD + n];
                b.y = Wl[kb * HD + n];
                acc = __builtin_amdgcn_wmma_f32_16x16x4_f32(
                        false, a, false, b, (short)0, acc, false, false);
            }
            // D layout: lane L, VGPR v -> D[(L<16?0:8)+v, L&15]
            if (lane < 16) {
                const int j = j0 + lane;
                uo[j] = acc[0];                                   // row 0: W @ Wl
                co[j] = acc[1] + ((j < HD) ? bl[j] : 0.0f);       // row 1: b @ Wl + bl
            }
        }
    }

    // softmax over att[0..11]
    float v  = (lane < PP) ? att[lane] : -1e30f;
    float mx = v;
    for (int o = 16; o > 0; o >>= 1) mx = fmaxf(mx, __shfl_xor(mx, o, 32));
    float e = (lane < PP) ? __expf(v - mx) : 0.0f;
    float s = e;
    for (int o = 16; o > 0; o >>= 1) s += __shfl_xor(s, o, 32);
    if (lane < PP) probs[lane] = e / s;
}

// ---------------------------------------------------------------------------
// K3: degree accumulation  deg[dst] += w[e]
// ---------------------------------------------------------------------------
__global__ void k_degree(const int* __restrict__ ei, const float* __restrict__ w,
                         float* __restrict__ deg, int E) {
    int e = blockIdx.x * blockDim.x + threadIdx.x;
    if (e >= E) return;
    int d = ei[E + e];
    __hip_atomic_fetch_add(&deg[d], w[e], __ATOMIC_RELAXED, __HIP_MEMORY_SCOPE_AGENT);
}

// ---------------------------------------------------------------------------
// K4: deg -> deg^-0.5 in place
// ---------------------------------------------------------------------------
__global__ void k_dinv(float* __restrict__ deg, int N) {
    int i = blockIdx.x * blockDim.x + threadIdx.x;
    if (i >= N) return;
    float d = deg[i];
    deg[i] = (d > 0.0f) ? (1.0f / sqrtf(d)) : 0.0f;
}

// ---------------------------------------------------------------------------
// K5: edge scatter  agg[dst,p] += dinv[s]*w*dinv[d] * x[s,p]
// ---------------------------------------------------------------------------
__global__ void k_scatter(const int* __restrict__ ei, const float* __restrict__ w,
                          const float* __restrict__ x, const float* __restrict__ dinv,
                          float* __restrict__ agg, int E) {
    int e = blockIdx.x * blockDim.x + threadIdx.x;
    if (e >= E) return;
    int s = ei[e];
    int d = ei[E + e];
    float nrm = dinv[s] * w[e] * dinv[d];
    const float* xs = x + (long)s * PP;
    float*       ad = agg + (long)d * PP;
#pragma unroll
    for (int p = 0; p < PP; ++p)
        __hip_atomic_fetch_add(&ad[p], nrm * xs[p],
                               __ATOMIC_RELAXED, __HIP_MEMORY_SCOPE_AGENT);
}

// ---------------------------------------------------------------------------
// K6: one wave per node. Self-loop term folded in here (agg + dinv^2 * x).
//   out[n] = b_out + sum_h Wout[h] * relu( sum_p probs[p]*(1-sigm)*tanh )
// ---------------------------------------------------------------------------
__global__ void k_final(const float* __restrict__ x, const float* __restrict__ agg,
                        const float* __restrict__ dinv,
                        const float* __restrict__ uz, const float* __restrict__ cz,
                        const float* __restrict__ uh, const float* __restrict__ ch,
                        const float* __restrict__ probs,
                        const float* __restrict__ Wout, const float* __restrict__ bout,
                        float* __restrict__ out, int N) {
    int wid  = (blockIdx.x * blockDim.x + threadIdx.x) >> 5;
    int lane = threadIdx.x & 31;
    if (wid >= N) return;

    float di = dinv[wid];
    float d2 = di * di;
    float a[PP], pr[PP];
#pragma unroll
    for (int p = 0; p < PP; ++p) {
        a[p]  = agg[(long)wid * PP + p] + d2 * x[(long)wid * PP + p];
        pr[p] = probs[p];
    }

    float outacc = 0.0f;
    for (int h = lane; h < HD; h += 32) {
        float U1 = uz[h], C1 = cz[h];
        float U2 = uh[h], C2 = ch[h];
        float hacc = 0.0f;
#pragma unroll
        for (int p = 0; p < PP; ++p) {
            float zg = 1.0f / (1.0f + __expf(-(a[p] * U1 + C1)));            // sigmoid
            float tg = 2.0f / (1.0f + __expf(-2.0f * (a[p] * U2 + C2))) - 1.0f; // tanh
            hacc += pr[p] * (1.0f - zg) * tg;
        }
        outacc += Wout[h] * fmaxf(hacc, 0.0f);
    }
    for (int o = 16; o > 0; o >>= 1) outacc += __shfl_xor(outacc, o, 32);
    if (lane == 0) out[wid] = outacc + bout[0];
}

// ---------------------------------------------------------------------------
extern "C" void kernel_launch(void* const* d_in, const int* in_sizes, int n_in,
                              void* d_out, int out_size, void* d_ws, size_t ws_size,
                              hipStream_t stream) {
    const float* x    = (const float*)d_in[0];
    const int*   ei   = (const int*)  d_in[1];
    const float* ew   = (const float*)d_in[2];
    const float* Wz   = (const float*)d_in[3];
    const float* bz   = (const float*)d_in[4];
    // d_in[5], d_in[6] = W_r, b_r  (algebraically dead: H0*R == 0)
    const float* Wh   = (const float*)d_in[7];
    const float* bh   = (const float*)d_in[8];
    const float* Wlz  = (const float*)d_in[9];
    const float* blz  = (const float*)d_in[10];
    // d_in[11], d_in[12] = Wl_r, bl_r (dead)
    const float* Wlh  = (const float*)d_in[13];
    const float* blh  = (const float*)d_in[14];
    const float* att  = (const float*)d_in[15];
    const float* Wout = (const float*)d_in[16];
    const float* bout = (const float*)d_in[17];
    float*       out  = (float*)d_out;

    const int N = in_sizes[0] / PP;
    const int E = in_sizes[2];

    // workspace layout (floats)
    float* ws    = (float*)d_ws;
    float* agg   = ws;                       // N*PP
    float* deg   = agg + (size_t)N * PP;     // N  (becomes dinv in place)
    float* uz    = deg + N;                  // 128 (valid j: 0..111, used: <100)
    float* cz    = uz + 128;
    float* uh    = cz + 128;
    float* ch    = uh + 128;
    float* probs = ch + 128;                 // 16

    const int BT = 256;

    k_init<<<(N * PP + BT - 1) / BT, BT, 0, stream>>>(agg, deg, N);

    k_precompute<<<1, 32, 0, stream>>>(Wz, bz, Wh, bh, Wlz, blz, Wlh, blh, att,
                                       uz, cz, uh, ch, probs);

    k_degree<<<(E + BT - 1) / BT, BT, 0, stream>>>(ei, ew, deg, E);

    k_dinv<<<(N + BT - 1) / BT, BT, 0, stream>>>(deg, N);

    k_scatter<<<(E + BT - 1) / BT, BT, 0, stream>>>(ei, ew, x, deg, agg, E);

    int waves_threads = N * 32;
    k_final<<<(waves_threads + BT - 1) / BT, BT, 0, stream>>>(
        x, agg, deg, uz, cz, uh, ch, probs, Wout, bout, out, N);
}